// GroupQueryAttention_31628139168153
// MI455X (gfx1250) — compile-verified
//
#include <hip/hip_runtime.h>
#include <hip/hip_bf16.h>

// ---------------------------------------------------------------------------
// GQA forward for MI455X (gfx1250), fp32 end-to-end via V_WMMA_F32_16X16X4_F32.
// Pipeline:
//   1. qp = q@Wq+bq   [4096,1024]      (gemm)
//   2. kp = k@Wk+bk   [4096,256]       (gemm)
//   3. vp = v@Wv+bv   [4096,256]       (gemm)
//   4. attn_raw = (Qh @ Kh^T) * 0.125  (batched gemm, transB, 32 batches)
//   5. softmax rows of attn (in place, in d_out attn region)
//   6. ctx = attn @ Vh                 (batched gemm, 32 batches)
//   7. out = ctx@Wo+bo                 (gemm)
// ---------------------------------------------------------------------------

typedef float v2f __attribute__((ext_vector_type(2)));
typedef float v8f __attribute__((ext_vector_type(8)));

#define DM   1024
#define HH   16
#define KVH  4
#define DK   64
#define KVD  256
#define BB   2
#define SS   2048

// Generic batched fp32 WMMA GEMM:
//   C[z] = alpha * A[z] (M x K) @ B[z] (K x N, or N x K if TRANSB) + bias
// Batch z decodes as b = z>>4, h = z&15, kv = h>>2 (GQA head grouping);
// each operand gets element-strides for (b, h, kv), applied to the uniform
// base pointer (stays in SGPRs). All intra-batch offsets are int32 so loads
// lower to saddr + 32-bit voffset instead of 64-bit VGPR pointer arithmetic.
//
// Block: 128 threads = 4 waves. Each wave computes a 32x64 strip of C
// (2 A-fragments x 4 B-fragments = 8 WMMAs per k-step of 4), with explicit
// register double-buffering of the fragments across k-steps.
// Requires: M % 128 == 0, N % 64 == 0, K % 8 == 0 (true for all uses here).
template <int TRANSB>
__global__ __launch_bounds__(128) void gqa_gemm_f32(
    const float* __restrict__ A, int lda,
    long long sAb, long long sAh, long long sAkv,
    const float* __restrict__ Bm, int ldb,
    long long sBb, long long sBh, long long sBkv,
    float* __restrict__ C, int ldc,
    long long sCb, long long sCh, long long sCkv,
    const float* __restrict__ bias,
    int K, float alpha)
{
    const int lane = threadIdx.x & 31;
    const int wave = threadIdx.x >> 5;
    const int lr   = lane & 15;   // 0..15  (row within A-frag / col within B-frag)
    const int hi   = lane >> 4;   // 0 or 1 (selects K pair / row-half)

    const int z  = blockIdx.z;
    const int b  = z >> 4;
    const int h  = z & 15;
    const int kv = h >> 2;
    A  += (long long)b * sAb + (long long)h * sAh + (long long)kv * sAkv;
    Bm += (long long)b * sBb + (long long)h * sBh + (long long)kv * sBkv;
    C  += (long long)b * sCb + (long long)h * sCh + (long long)kv * sCkv;

    const int row0 = blockIdx.y * 128 + wave * 32;
    const int col0 = blockIdx.x * 64;

    // Int32 element offsets (uniform base stays scalar).
    // A frag m: a.x = A[row + k + 2*hi], a.y = +1  (one 8-byte load)
    const int aoff0 = (row0 + lr) * lda + 2 * hi;
    const int aoff1 = aoff0 + 16 * lda;

    int boff[4];
#pragma unroll
    for (int t = 0; t < 4; ++t) {
        if (TRANSB) {
            // B given as N x K row-major: frag.x/.y = B[col][k+2*hi .. +1] (contiguous)
            boff[t] = (col0 + 16 * t + lr) * ldb + 2 * hi;
        } else {
            // B is K x N row-major: frag.x = B[k+2*hi][col], frag.y = B[k+2*hi+1][col]
            boff[t] = (2 * hi) * ldb + (col0 + 16 * t + lr);
        }
    }

    v8f acc[2][4] = {{v8f{}, v8f{}, v8f{}, v8f{}},
                     {v8f{}, v8f{}, v8f{}, v8f{}}};

    // ---- fragment loaders -------------------------------------------------
    auto ldA0 = [&](int k) { return *(const v2f*)(A + aoff0 + k); };
    auto ldA1 = [&](int k) { return *(const v2f*)(A + aoff1 + k); };
    auto ldB  = [&](int t, int k) {
        v2f bb;
        if (TRANSB) {
            bb = *(const v2f*)(Bm + boff[t] + k);
        } else {
            bb.x = Bm[boff[t] + k * ldb];
            bb.y = Bm[boff[t] + k * ldb + ldb];
        }
        return bb;
    };

    // ---- software-pipelined k loop (double-buffered fragments) ------------
    v2f a0c = ldA0(0);
    v2f a1c = ldA1(0);
    v2f bc[4];
#pragma unroll
    for (int t = 0; t < 4; ++t) bc[t] = ldB(t, 0);

#pragma unroll 2
    for (int k = 0; k < K - 4; k += 4) {
        // issue next-iteration loads first so WMMAs below only wait on the
        // previous iteration's loads (full-iteration latency slack)
        v2f a0n = ldA0(k + 4);
        v2f a1n = ldA1(k + 4);
        v2f bn[4];
#pragma unroll
        for (int t = 0; t < 4; ++t) bn[t] = ldB(t, k + 4);

#pragma unroll
        for (int t = 0; t < 4; ++t) {
            acc[0][t] = __builtin_amdgcn_wmma_f32_16x16x4_f32(
                false, a0c, false, bc[t], (short)0, acc[0][t], false, false);
            acc[1][t] = __builtin_amdgcn_wmma_f32_16x16x4_f32(
                false, a1c, false, bc[t], (short)0, acc[1][t], false, false);
        }
        a0c = a0n;
        a1c = a1n;
#pragma unroll
        for (int t = 0; t < 4; ++t) bc[t] = bn[t];
    }
#pragma unroll
    for (int t = 0; t < 4; ++t) {
        acc[0][t] = __builtin_amdgcn_wmma_f32_16x16x4_f32(
            false, a0c, false, bc[t], (short)0, acc[0][t], false, false);
        acc[1][t] = __builtin_amdgcn_wmma_f32_16x16x4_f32(
            false, a1c, false, bc[t], (short)0, acc[1][t], false, false);
    }

    // ---- store: C/D layout — VGPR r holds row (row0 + m*16 + hi*8 + r) ----
#pragma unroll
    for (int t = 0; t < 4; ++t) {
        const int n  = col0 + 16 * t + lr;
        const float bv = bias ? bias[n] : 0.0f;
#pragma unroll
        for (int m = 0; m < 2; ++m) {
            const int crow = (row0 + m * 16 + hi * 8) * ldc + n;
#pragma unroll
            for (int r = 0; r < 8; ++r) {
                C[crow + r * ldc] = alpha * acc[m][t][r] + bv;
            }
        }
    }
}

// Row softmax, in place. One 256-thread block per row of `ncols` floats.
__global__ __launch_bounds__(256) void gqa_softmax_rows(
    float* __restrict__ attn, int ncols)
{
    float* p = attn + (long long)blockIdx.x * ncols;
    __shared__ float red[256];
    const int tid = threadIdx.x;

    float m = -3.0e38f;
    for (int c = tid; c < ncols; c += 256) m = fmaxf(m, p[c]);
    red[tid] = m;
    __syncthreads();
#pragma unroll
    for (int s = 128; s > 0; s >>= 1) {
        if (tid < s) red[tid] = fmaxf(red[tid], red[tid + s]);
        __syncthreads();
    }
    m = red[0];
    __syncthreads();

    float sum = 0.0f;
    for (int c = tid; c < ncols; c += 256) {
        float e = __expf(p[c] - m);
        p[c] = e;
        sum += e;
    }
    red[tid] = sum;
    __syncthreads();
#pragma unroll
    for (int s = 128; s > 0; s >>= 1) {
        if (tid < s) red[tid] += red[tid + s];
        __syncthreads();
    }
    const float inv = 1.0f / red[0];
    for (int c = tid; c < ncols; c += 256) p[c] *= inv;
}

extern "C" void kernel_launch(void* const* d_in, const int* in_sizes, int n_in,
                              void* d_out, int out_size, void* d_ws, size_t ws_size,
                              hipStream_t stream) {
    (void)in_sizes; (void)n_in; (void)out_size; (void)ws_size;

    const float* q  = (const float*)d_in[0];
    const float* k  = (const float*)d_in[1];
    const float* v  = (const float*)d_in[2];
    const float* Wq = (const float*)d_in[3];
    const float* bq = (const float*)d_in[4];
    const float* Wk = (const float*)d_in[5];
    const float* bk = (const float*)d_in[6];
    const float* Wv = (const float*)d_in[7];
    const float* bv = (const float*)d_in[8];
    const float* Wo = (const float*)d_in[9];
    const float* bo = (const float*)d_in[10];

    float* out  = (float*)d_out;                       // [B,S,1024]
    float* attn = out + (long long)BB * SS * DM;       // [B,H,S,S]

    const long long M = (long long)BB * SS;            // 4096
    float* qp  = (float*)d_ws;                         // [4096,1024]
    float* kp  = qp + M * DM;                          // [4096,256]
    float* vp  = kp + M * KVD;                         // [4096,256]
    float* ctx = vp + M * KVD;                         // [4096,1024]

    dim3 blk(128);

    // 1-3: input projections (single batch, zero batch strides)
    gqa_gemm_f32<0><<<dim3(DM / 64, M / 128, 1), blk, 0, stream>>>(
        q, DM, 0, 0, 0, Wq, DM, 0, 0, 0, qp, DM, 0, 0, 0, bq, DM, 1.0f);
    gqa_gemm_f32<0><<<dim3(KVD / 64, M / 128, 1), blk, 0, stream>>>(
        k, DM, 0, 0, 0, Wk, KVD, 0, 0, 0, kp, KVD, 0, 0, 0, bk, DM, 1.0f);
    gqa_gemm_f32<0><<<dim3(KVD / 64, M / 128, 1), blk, 0, stream>>>(
        v, DM, 0, 0, 0, Wv, KVD, 0, 0, 0, vp, KVD, 0, 0, 0, bv, DM, 1.0f);

    // 4: scores = Qh @ Kh^T * (1/sqrt(dk)); 32 batches (b,h), kv = h>>2
    gqa_gemm_f32<1><<<dim3(SS / 64, SS / 128, BB * HH), blk, 0, stream>>>(
        qp, DM, (long long)SS * DM, DK, 0,             // A: Q head h at col h*64
        kp, KVD, (long long)SS * KVD, 0, DK,           // B: K head kv at col kv*64 ([S,dk] row-major -> transB)
        attn, SS, (long long)HH * SS * SS, (long long)SS * SS, 0,
        nullptr, DK, 0.125f);

    // 5: softmax over the last dim of attn (B*H*S rows of S)
    gqa_softmax_rows<<<dim3(BB * HH * SS), dim3(256), 0, stream>>>(attn, SS);

    // 6: ctx = attn @ Vh ; write head h into columns h*64 of [B,S,1024]
    gqa_gemm_f32<0><<<dim3(DK / 64, SS / 128, BB * HH), blk, 0, stream>>>(
        attn, SS, (long long)HH * SS * SS, (long long)SS * SS, 0,
        vp, KVD, (long long)SS * KVD, 0, DK,
        ctx, DM, (long long)SS * DM, DK, 0,
        nullptr, SS, 1.0f);

    // 7: out = ctx @ Wo + bo
    gqa_gemm_f32<0><<<dim3(DM / 64, M / 128, 1), blk, 0, stream>>>(
        ctx, DM, 0, 0, 0, Wo, DM, 0, 0, 0, out, DM, 0, 0, 0, bo, DM, 1.0f);
}